// PureKernalMetricLogits_53163105190628
// MI455X (gfx1250) — compile-verified
//
#include <hip/hip_runtime.h>
#include <hip/hip_bf16.h>

typedef float v2f __attribute__((ext_vector_type(2)));
typedef float v8f __attribute__((ext_vector_type(8)));

#define FEAT_DIM   128
#define CLASS_NUM  10000
#define BATCH      4096
#define LDF        132          // LDS row stride in floats (128 + 4 pad -> conflict-free)
#define BT         64           // feat rows per block
#define CT         16           // weight cols per block
#define INV_T      (1.0f / 1.8f)

// ---------------------------------------------------------------------------
// Kernel 1: squared row norms for feat (4096 rows) and weights (10000 rows).
// One wave32 per row; float4 loads; shuffle reduction. Writes f2[] and w2[].
// ---------------------------------------------------------------------------
__global__ __launch_bounds__(256) void sqnorm_kernel(const float* __restrict__ feat,
                                                     const float* __restrict__ wts,
                                                     float* __restrict__ f2,
                                                     float* __restrict__ w2) {
    const int wid  = (blockIdx.x * blockDim.x + threadIdx.x) >> 5;
    const int lane = threadIdx.x & 31;
    if (wid >= BATCH + CLASS_NUM) return;

    const float* src;
    float* dst;
    int row;
    if (wid < BATCH) { src = feat; dst = f2; row = wid; }
    else             { src = wts;  dst = w2; row = wid - BATCH; }

    const float4 v = ((const float4*)(src + (size_t)row * FEAT_DIM))[lane];
    float s = v.x * v.x + v.y * v.y + v.z * v.z + v.w * v.w;
    #pragma unroll
    for (int off = 16; off > 0; off >>= 1) s += __shfl_xor(s, off, 32);
    if (lane == 0) dst[row] = s;
}

// ---------------------------------------------------------------------------
// Kernel 2: per-sample true-class kernel value cor[b] = exp(-relu(d)/1.8).
// One wave32 per sample; gathers W[label[b]].
// ---------------------------------------------------------------------------
__global__ __launch_bounds__(256) void cor_kernel(const float* __restrict__ feat,
                                                  const float* __restrict__ wts,
                                                  const int*   __restrict__ label,
                                                  const float* __restrict__ f2,
                                                  const float* __restrict__ w2,
                                                  float* __restrict__ cor) {
    const int b    = (blockIdx.x * blockDim.x + threadIdx.x) >> 5;
    const int lane = threadIdx.x & 31;
    if (b >= BATCH) return;

    const int c = label[b];
    const float4 fv = ((const float4*)(feat + (size_t)b * FEAT_DIM))[lane];
    const float4 wv = ((const float4*)(wts  + (size_t)c * FEAT_DIM))[lane];
    float s = fv.x * wv.x + fv.y * wv.y + fv.z * wv.z + fv.w * wv.w;
    #pragma unroll
    for (int off = 16; off > 0; off >>= 1) s += __shfl_xor(s, off, 32);
    if (lane == 0) {
        float m = fmaxf(f2[b] + w2[c] - 2.0f * s, 0.0f);
        cor[b] = __expf(-m * INV_T);
    }
}

// ---------------------------------------------------------------------------
// Kernel 3: fixed-order reduction of cor[] -> scale = log(C-1)/max(mean,0.5).
// Single block => deterministic.
// ---------------------------------------------------------------------------
__global__ __launch_bounds__(256) void scale_kernel(const float* __restrict__ cor,
                                                    float* __restrict__ scale) {
    __shared__ float red[256];
    float s = 0.0f;
    for (int i = threadIdx.x; i < BATCH; i += 256) s += cor[i];
    red[threadIdx.x] = s;
    __syncthreads();
    #pragma unroll
    for (int off = 128; off > 0; off >>= 1) {
        if ((int)threadIdx.x < off) red[threadIdx.x] += red[threadIdx.x + off];
        __syncthreads();
    }
    if (threadIdx.x == 0) {
        float avg = fmaxf(red[0] / (float)BATCH, 0.5f);
        scale[0] = __logf((float)CLASS_NUM - 1.0f) / avg;
    }
}

// ---------------------------------------------------------------------------
// Kernel 4: main GEMM + fused epilogue.
// Block = 128 threads (4 waves). LDS: feat tile 64x128 (stride 132) + W tile
// 16x128. Each wave computes a 16x16 tile of out via 32x V_WMMA_F32_16X16X4_F32.
//
// Fragment layouts (ISA 7.12.2, wave32):
//   A (16x4 f32): lane l -> row (l&15); VGPR0/1 = K = 2*(l>>4), 2*(l>>4)+1
//   B (4x16 f32): lane l -> col (l&15); VGPR0/1 = K = 2*(l>>4), 2*(l>>4)+1
//   D (16x16 f32, 8 VGPRs): VGPR v, lane l -> M = v + 8*(l>>4), N = (l&15)
// ---------------------------------------------------------------------------
__global__ __launch_bounds__(128) void gemm_epilogue_kernel(const float* __restrict__ feat,
                                                            const float* __restrict__ wts,
                                                            const float* __restrict__ f2,
                                                            const float* __restrict__ w2,
                                                            const float* __restrict__ scaleP,
                                                            float* __restrict__ out) {
    __shared__ __align__(16) float sF[BT * LDF];
    __shared__ __align__(16) float sW[CT * LDF];

    const int tid   = threadIdx.x;
    const int lane  = tid & 31;
    const int wave  = tid >> 5;
    const int cBase = blockIdx.x * CT;    // 625 tiles * 16 = 10000
    const int bBase = blockIdx.y * BT;    // 64 tiles * 64  = 4096

    // Stage feat tile: 64 rows x 128 cols = 2048 float4, 16 per thread.
    for (int i = tid; i < BT * (FEAT_DIM / 4); i += 128) {
        const int r = i >> 5, c4 = i & 31;
        float4 v = ((const float4*)(feat + (size_t)(bBase + r) * FEAT_DIM))[c4];
        *(float4*)(&sF[r * LDF + c4 * 4]) = v;
    }
    // Stage weights tile: 16 rows x 128 cols = 512 float4, 4 per thread.
    for (int i = tid; i < CT * (FEAT_DIM / 4); i += 128) {
        const int r = i >> 5, c4 = i & 31;
        float4 v = ((const float4*)(wts + (size_t)(cBase + r) * FEAT_DIM))[c4];
        *(float4*)(&sW[r * LDF + c4 * 4]) = v;
    }
    __syncthreads();

    const int half = lane >> 4;     // 0 / 1
    const int lr   = lane & 15;
    const int koff = half * 2;      // K sub-offset for this lane's half

    const float* aRow = &sF[(wave * CT + lr) * LDF + koff];  // A: row = lr
    const float* bRow = &sW[lr * LDF + koff];                // B: col = lr

    v8f acc = {};
    #pragma unroll
    for (int k = 0; k < FEAT_DIM; k += 4) {
        v2f a = *(const v2f*)(aRow + k);   // ds_load_b64
        v2f b = *(const v2f*)(bRow + k);   // ds_load_b64
        acc = __builtin_amdgcn_wmma_f32_16x16x4_f32(
            /*neg_a=*/false, a, /*neg_b=*/false, b,
            /*c_mod=*/(short)0, acc, /*reuse_a=*/false, /*reuse_b=*/false);
    }

    const float scale = scaleP[0];
    const int   col   = cBase + lr;
    const float w2c   = w2[col];

    #pragma unroll
    for (int v = 0; v < 8; ++v) {
        const int row = bBase + wave * CT + v + 8 * half;
        const float m = fmaxf(f2[row] + w2c - 2.0f * acc[v], 0.0f);
        out[(size_t)row * CLASS_NUM + col] = scale * __expf(-m * INV_T);
    }
}

// ---------------------------------------------------------------------------
extern "C" void kernel_launch(void* const* d_in, const int* in_sizes, int n_in,
                              void* d_out, int out_size, void* d_ws, size_t ws_size,
                              hipStream_t stream) {
    const float* feat  = (const float*)d_in[0];   // [4096,128] f32
    const float* wts   = (const float*)d_in[1];   // [10000,128] f32
    const int*   label = (const int*)d_in[2];     // [4096]
    float*       out   = (float*)d_out;           // [4096,10000] f32

    // Workspace layout (floats): [0]=scale, then f2, w2, cor.
    float* ws_f  = (float*)d_ws;
    float* scale = ws_f;
    float* f2    = ws_f + 16;
    float* w2    = f2 + BATCH;
    float* cor   = w2 + CLASS_NUM;

    // 1) row norms: one wave per row, (4096+10000) rows, 8 waves/block.
    {
        const int waves = BATCH + CLASS_NUM;
        const int blocks = (waves + 7) / 8;
        sqnorm_kernel<<<blocks, 256, 0, stream>>>(feat, wts, f2, w2);
    }
    // 2) per-sample true-class kernel value.
    {
        const int blocks = (BATCH + 7) / 8;
        cor_kernel<<<blocks, 256, 0, stream>>>(feat, wts, label, f2, w2, cor);
    }
    // 3) deterministic mean -> scale.
    scale_kernel<<<1, 256, 0, stream>>>(cor, scale);

    // 4) WMMA GEMM + fused exp epilogue. 625 x 64 blocks of 128 threads.
    {
        dim3 grid(CLASS_NUM / CT, BATCH / BT);
        gemm_epilogue_kernel<<<grid, 128, 0, stream>>>(feat, wts, f2, w2, scale, out);
    }
}